// UNetEncoderBlock_23484881174876
// MI455X (gfx1250) — compile-verified
//
#include <hip/hip_runtime.h>

// MI455X (gfx1250) fused kernel for: conv(1->64, 3x3, SAME) -> depthwise conv(64g, 3x3, SAME) -> ReLU
//
// Roofline: output write is 512MB; min traffic ~520MB -> ~22us @ 23.3TB/s.
// Fusing both convs through LDS avoids a 1GB round-trip of the intermediate.
// conv1 is a (M=64ch, K=9taps, N=pixels) fp32 GEMM -> V_WMMA_F32_16X16X4_F32
// on the matrix pipe; conv2 (depthwise, not GEMM-shaped) runs on VALU and
// co-executes. conv2 is register-blocked 4-rows-per-thread so LDS reads drop
// from 9/output to 4.5/output, keeping LDS bandwidth under the HBM roofline.
// Border handling: out1 is explicitly zeroed outside the image before conv2
// consumes it (matches the reference's SAME zero-padding on both convs).

typedef __attribute__((ext_vector_type(2))) float v2f;
typedef __attribute__((ext_vector_type(8))) float v8f;

#define TILE   16     // 16x16 output pixels per workgroup (per batch)
#define R1H    18     // out1 halo region is 18x18
#define NPIX   324    // 18*18
#define OSTR   386    // out1 LDS per-channel stride (+8 chans -> +16 banks: halves of a wave stay conflict-free)
#define XCOLS  20
#define XSZ    512    // 20x20 real + zero slack so speculative tap-9..11 reads stay in-bounds

__global__ __launch_bounds__(256)
void unet_enc_fused(const float* __restrict__ x,
                    const float* __restrict__ w1,
                    const float* __restrict__ w2,
                    float* __restrict__ out)
{
    __shared__ float xs[XSZ];          // x0 tile, zero-padded
    __shared__ float w1s[64*9 + 8];    // +pad: speculative reads of tap 9..11 stay in-bounds
    __shared__ float w2s[64*9 + 8];
    __shared__ float o1s[16*OSTR];     // out1 for one 16-channel group

    const int t    = threadIdx.x;
    const int lane = t & 31;
    const int wv   = t >> 5;           // wave id, 8 waves of 32 (wave32)
    const int b    = blockIdx.z;
    const int X0   = blockIdx.x * TILE;
    const int Y0   = blockIdx.y * TILE;

    const float* xb = x + (size_t)b * 3 * 512 * 512;   // channel 0 plane of batch b

    // ---- Phase 0: load x0 halo tile (zero-padded) + both weight tensors ----
    for (int idx = t; idx < XSZ; idx += 256) {
        int r = idx / XCOLS, c = idx - r * XCOLS;
        int gy = Y0 - 2 + r, gx = X0 - 2 + c;
        float v = 0.f;
        if (r < 20 && gy >= 0 && gy < 512 && gx >= 0 && gx < 512)
            v = xb[gy * 512 + gx];
        xs[idx] = v;
    }
    for (int idx = t; idx < 576; idx += 256) {
        w1s[idx] = w1[idx];
        w2s[idx] = w2[idx];
    }
    __syncthreads();

    const int lm    = lane & 15;              // A: channel-in-group / B: pixel-in-tile
    const int khalf = (lane < 16) ? 0 : 2;    // K position within a 4-chunk

    // conv2 thread mapping: 16 cols x 4 row-quads x 4 channel-slots
    const int tx = t & 15;
    const int tq = (t >> 4) & 3;       // rows tq*4 .. tq*4+3
    const int c4 = t >> 6;             // channel slot 0..3

    for (int cg = 0; cg < 4; ++cg) {
        // ---- A fragments: w1 for this 16-channel group, K padded 9 -> 12 ----
        float a0[3], a1[3];
        {
            const float* wrow = &w1s[(cg * 16 + lm) * 9];
            #pragma unroll
            for (int kc = 0; kc < 3; ++kc) {
                int t0 = kc * 4 + khalf;
                a0[kc] = (t0     < 9) ? wrow[t0]     : 0.f;
                a1[kc] = (t0 + 1 < 9) ? wrow[t0 + 1] : 0.f;
            }
        }

        // ---- conv1 via WMMA: each wave computes 3 N-tiles of 16 pixels ----
        #pragma unroll
        for (int k = 0; k < 3; ++k) {
            int nt = wv * 3 + k;            // 0..23, wave-uniform
            int q  = nt * 16 + lm;          // pixel index in padded 384
            int qy = q / 18, qx = q - qy * 18;

            v8f acc = {};
            #pragma unroll
            for (int kc = 0; kc < 3; ++kc) {
                int t0 = kc * 4 + khalf;
                int t1 = t0 + 1;
                v2f A; A.x = a0[kc]; A.y = a1[kc];
                // B: patch values; taps 9..11 read in-bounds xs, but A is 0 for them
                v2f B;
                B.x = xs[(qy + t0 / 3) * XCOLS + qx + (t0 - (t0 / 3) * 3)];
                B.y = xs[(qy + t1 / 3) * XCOLS + qx + (t1 - (t1 / 3) * 3)];
                acc = __builtin_amdgcn_wmma_f32_16x16x4_f32(
                        false, A, false, B, (short)0, acc, false, false);
            }

            // D -> LDS; zero positions outside image (conv2 SAME zero-padding)
            int gy = Y0 - 1 + qy, gx = X0 - 1 + qx;
            bool valid = (q < NPIX) && (gy >= 0) && (gy < 512) && (gx >= 0) && (gx < 512);
            int chbase = (lane < 16) ? 0 : 8;   // C/D layout: VGPR r -> rows r / r+8
            #pragma unroll
            for (int r = 0; r < 8; ++r)
                o1s[(chbase + r) * OSTR + q] = valid ? acc[r] : 0.f;
        }
        __syncthreads();

        // ---- conv2 depthwise + ReLU, register-blocked 4 rows per thread ----
        // Thread (tx, tq, c4) computes out rows tq*4..tq*4+3, column tx, for
        // channels c4, 4+c4, 8+c4, 12+c4 of this group. Needs out1 region
        // rows tq*4..tq*4+5, cols tx..tx+2 -> 18 LDS reads per 4 outputs.
        #pragma unroll
        for (int cc = 0; cc < 4; ++cc) {
            int c = cc * 4 + c4;
            const float* wrow = &w2s[(cg * 16 + c) * 9];
            const float* src  = &o1s[c * OSTR];
            float acc0 = 0.f, acc1 = 0.f, acc2 = 0.f, acc3 = 0.f;
            #pragma unroll
            for (int rr = 0; rr < 6; ++rr) {
                int py = tq * 4 + rr;
                float v0 = src[py * R1H + tx + 0];
                float v1 = src[py * R1H + tx + 1];
                float v2 = src[py * R1H + tx + 2];
                #pragma unroll
                for (int di = 0; di < 3; ++di) {
                    int r = rr - di;          // output row in quad fed by this region row
                    if (r >= 0 && r < 4) {
                        float s = fmaf(wrow[di * 3 + 0], v0,
                                  fmaf(wrow[di * 3 + 1], v1,
                                       wrow[di * 3 + 2] * v2));
                        if (r == 0) acc0 += s;
                        if (r == 1) acc1 += s;
                        if (r == 2) acc2 += s;
                        if (r == 3) acc3 += s;
                    }
                }
            }
            size_t base = (((size_t)b * 64 + cg * 16 + c) * 512 + (Y0 + tq * 4)) * 512 + (X0 + tx);
            // Output is write-once streaming data: keep it out of L2 (NT hint)
            __builtin_nontemporal_store(fmaxf(acc0, 0.f), &out[base          ]);
            __builtin_nontemporal_store(fmaxf(acc1, 0.f), &out[base + 512    ]);
            __builtin_nontemporal_store(fmaxf(acc2, 0.f), &out[base + 1024   ]);
            __builtin_nontemporal_store(fmaxf(acc3, 0.f), &out[base + 1536   ]);
        }
        __syncthreads();   // o1s reused by next channel group
    }
}

extern "C" void kernel_launch(void* const* d_in, const int* in_sizes, int n_in,
                              void* d_out, int out_size, void* d_ws, size_t ws_size,
                              hipStream_t stream) {
    const float* x  = (const float*)d_in[0];   // (8,3,512,512)
    const float* w1 = (const float*)d_in[1];   // (64,1,3,3)
    const float* w2 = (const float*)d_in[2];   // (64,1,3,3)
    float* out = (float*)d_out;                // (8,64,512,512)

    dim3 grid(512 / TILE, 512 / TILE, 8);
    unet_enc_fused<<<grid, 256, 0, stream>>>(x, w1, w2, out);
}